// SSL2d_70454643524031
// MI455X (gfx1250) — compile-verified
//
#include <hip/hip_runtime.h>
#include <hip/hip_bf16.h>

#define HDIM 56
#define WDIM 56
#define PLANE (HDIM * WDIM)   // 3136 floats = 12544 bytes per (n,c) plane
#define NCHUNK (PLANE / 4)    // 784 float4 chunks; W%4==0 so chunks never cross rows
#define NCH 384
#define NB 32
#define TPB 256               // 8 waves (wave32)

typedef unsigned int v4u __attribute__((ext_vector_type(4)));
typedef int          v4i __attribute__((ext_vector_type(4)));
typedef int          v8i __attribute__((ext_vector_type(8)));

__global__ __launch_bounds__(TPB) void ssl2d_tdm_kernel(
    const float* __restrict__ x,
    const float* __restrict__ a,
    const float* __restrict__ b,
    float* __restrict__ out) {
  __shared__ float tile[PLANE];

  const int plane_id = blockIdx.x;            // n * C + c
  const int c = plane_id % NCH;
  const long long base = (long long)plane_id * PLANE;
  const float* __restrict__ xp = x + base;
  float* __restrict__ op = out + base;
  const int tid = threadIdx.x;

#if __has_builtin(__builtin_amdgcn_tensor_load_to_lds) && __has_builtin(__builtin_amdgcn_s_wait_tensorcnt)
  // --- Stage the whole plane into LDS with one TDM DMA (wave 0 only).
  // TDM ignores EXEC and issues once per executing wave, so the branch must
  // exclude entire waves, not lanes.
  if ((tid >> 5) == 0) {
    const unsigned long long ga = (unsigned long long)(const void*)xp;
    const unsigned lds_off = (unsigned)(unsigned long long)(const void*)&tile[0];

    // D# group 0 (128b): count=1 | lds_addr | global_addr[56:0] | type=2
    v4u g0;
    g0[0] = 1u;                                           // count=1, is_restore=0, gather off
    g0[1] = lds_off;                                      // LDS byte address
    g0[2] = (unsigned)(ga & 0xFFFFFFFFull);               // global_addr[31:0]
    g0[3] = (unsigned)((ga >> 32) & 0x01FFFFFFull)        // global_addr[56:32]
            | (2u << 30);                                 // type = 2 ("image")

    // D# group 1 (256b): 1-D contiguous tile of 3136 x 4-byte elements.
    v8i g1 = {};
    g1[0] = (int)(2u << 16);                 // workgroup_mask=0, data_size=2 (4B)
    g1[1] = (int)((unsigned)PLANE << 16);    // tensor_dim0[15:0] @ bits[63:48]
    g1[2] = (int)(1u << 16);                 // tensor_dim0[31:16]=0, tensor_dim1=1
    g1[3] = (int)((unsigned)PLANE << 16);    // tensor_dim1 hi=0, tile_dim0=3136
    g1[4] = 0;                               // tile_dim1=0 (unused), tile_dim2=0
    g1[5] = (int)PLANE;                      // tensor_dim0_stride[31:0]
    g1[6] = 0;                               // stride hi / tensor_dim1_stride lo
    g1[7] = 0;

    v4i gz = {};                             // groups 2/3 unused (<=2D tensor)
    v8i gx = {};                             // extra operand on clang-23 lane (zero-filled)

    __builtin_amdgcn_tensor_load_to_lds(g0, g1, gz, gz, gx, 0);
    __builtin_amdgcn_s_wait_tensorcnt(0);    // s_wait_tensorcnt 0
  }
  __syncthreads();                           // publish LDS to all 8 waves
#else
  // Fallback: cooperative synchronous staging (still one HBM read per element).
  for (int i = tid; i < PLANE; i += TPB) tile[i] = xp[i];
  __syncthreads();
#endif

  // --- Per-channel constants (uniform across the block -> scalarized).
  const float av = a[c];
  const float bv = b[c];
  const float iaf = floorf(av);
  const float ibf = floorf(bv);
  const float fa = av - iaf;
  const float fb = bv - ibf;
  const int ia = (int)iaf;
  const int ib = (int)ibf;
  const float w00 = (1.0f - fa) * (1.0f - fb);
  const float w01 = (1.0f - fa) * fb;
  const float w10 = fa * (1.0f - fb);
  const float w11 = fa * fb;

  // --- Bilinear gather from LDS, 4 pixels (one row segment) per iteration.
  // 5-wide tap window shared between adjacent pixels: 10 LDS b32 reads per
  // 4 outputs. One b128 store per chunk; consecutive lanes -> consecutive
  // 16B addresses (fully coalesced).
  for (int q = tid; q < NCHUNK; q += TPB) {
    const int h = q / (WDIM / 4);            // q / 14
    const int w4 = (q - h * (WDIM / 4)) * 4; // first col of this chunk
    const int hy = h + ia;

    const bool vh0 = (hy >= 0) & (hy < HDIM);
    const bool vh1 = (hy + 1 >= 0) & (hy + 1 < HDIM);
    const int hy0 = min(max(hy, 0), HDIM - 1);
    const int hy1 = min(max(hy + 1, 0), HDIM - 1);
    const int row0 = hy0 * WDIM;
    const int row1 = hy1 * WDIM;

    float t0[5], t1[5];
#pragma unroll
    for (int k = 0; k < 5; ++k) {
      const int wx = w4 + ib + k;
      const bool vw = (wx >= 0) & (wx < WDIM);
      const int wxc = min(max(wx, 0), WDIM - 1);
      const float u0 = tile[row0 + wxc];
      const float u1 = tile[row1 + wxc];
      t0[k] = (vh0 & vw) ? u0 : 0.0f;
      t1[k] = (vh1 & vw) ? u1 : 0.0f;
    }

    float r[4];
#pragma unroll
    for (int k = 0; k < 4; ++k)
      r[k] = w00 * t0[k] + w01 * t0[k + 1] + w10 * t1[k] + w11 * t1[k + 1];

    *(float4*)(op + q * 4) = make_float4(r[0], r[1], r[2], r[3]);
  }
}

extern "C" void kernel_launch(void* const* d_in, const int* in_sizes, int n_in,
                              void* d_out, int out_size, void* d_ws, size_t ws_size,
                              hipStream_t stream) {
  const float* x = (const float*)d_in[0];
  const float* a = (const float*)d_in[1];
  const float* b = (const float*)d_in[2];
  float* out = (float*)d_out;

  const int nplanes = NB * NCH;  // 12288 workgroups, one 56x56 plane each
  ssl2d_tdm_kernel<<<dim3(nplanes), dim3(TPB), 0, stream>>>(x, a, b, out);
}